// Encoder_45183055954421
// MI455X (gfx1250) — compile-verified
//
#include <hip/hip_runtime.h>

// ---------------------------------------------------------------------------
// 2-layer GCN for MI455X (gfx1250, wave32).
//   memory-bound edge scatter -> fp32 atomics into L2-resident accumulators
//   dense x@W               -> V_WMMA_F32_16X16X4_F32 chained over K (exact fp32)
// ---------------------------------------------------------------------------

typedef __attribute__((ext_vector_type(2))) float v2f;
typedef __attribute__((ext_vector_type(8))) float v8f;

static constexpr int NN   = 100000;   // nodes
static constexpr int NE   = 1600000;  // edges (self-loops handled separately)
static constexpr int DIN  = 16;
static constexpr int DHID = 32;
static constexpr int DOUT = 16;

// ---------------------------------------------------------------- utilities
__global__ void k_zero_f32(float* __restrict__ p, int n) {
  int i = blockIdx.x * blockDim.x + threadIdx.x;
  if (i < n) p[i] = 0.0f;
}

__global__ void k_deg(const int* __restrict__ dst, int* __restrict__ deg) {
  int e = blockIdx.x * blockDim.x + threadIdx.x;
  if (e < NE) atomicAdd(&deg[dst[e]], 1);
}

__global__ void k_dinv(const int* __restrict__ deg, float* __restrict__ dinv) {
  int i = blockIdx.x * blockDim.x + threadIdx.x;
  if (i < NN) dinv[i] = rsqrtf((float)deg[i] + 1.0f);  // +1 self-loop => deg>=1
}

// --------------------------------------------------- dense transform (WMMA)
// One wave computes a 16-node tile of Y = X[idx] @ W using V_WMMA_F32_16X16X4_F32
// chained K/4 times. ISA layouts (CDNA5 ISA):
//   A 16x4 f32 : lane m=lane&15 holds row m; VGPR{0,1} = K {0,1} (lanes 0-15)
//                or K {2,3} (lanes 16-31)
//   B 4x16 f32 : symmetric, lane n=lane&15 is column n
//   C/D 16x16  : 8 VGPRs; VGPR r = row r (lanes 0-15) / row r+8 (lanes 16-31)
template<int K, int NOUT, bool GATHER>
__global__ void k_linear_wmma(const float* __restrict__ X,
                              const float* __restrict__ W,
                              const int*  __restrict__ xidx,
                              float* __restrict__ Y) {
  constexpr int NT = NOUT / 16;                 // output column tiles
  const int lane = threadIdx.x & 31;
  const int wave = threadIdx.x >> 5;
  const int tile = blockIdx.x * (blockDim.x >> 5) + wave;
  if (tile >= NN / 16) return;                  // uniform per wave (EXEC all-1s)

  const int m    = lane & 15;                   // row within tile / col within B
  const int half = lane >> 4;
  const int node = tile * 16 + m;
  const int row  = GATHER ? xidx[node] : node;
  const float* __restrict__ xrow = X + (long long)row * K;

  v8f acc[NT];
  const v8f vz = {0.f, 0.f, 0.f, 0.f, 0.f, 0.f, 0.f, 0.f};
#pragma unroll
  for (int t = 0; t < NT; ++t) acc[t] = vz;

#pragma unroll
  for (int c = 0; c < K / 4; ++c) {
    const int kb = 4 * c + 2 * half;            // this lane's K base
    v2f a = *(const v2f*)(xrow + kb);           // A[m][kb..kb+1], 8B aligned
#pragma unroll
    for (int t = 0; t < NT; ++t) {
      v2f b;
      b.x = W[(kb + 0) * NOUT + t * 16 + m];    // B[kb  ][n]
      b.y = W[(kb + 1) * NOUT + t * 16 + m];    // B[kb+1][n]
      acc[t] = __builtin_amdgcn_wmma_f32_16x16x4_f32(
          /*neg_a=*/false, a, /*neg_b=*/false, b,
          /*c_mod=*/(short)0, acc[t], /*reuse_a=*/false, /*reuse_b=*/false);
    }
  }

#pragma unroll
  for (int t = 0; t < NT; ++t) {
#pragma unroll
    for (int r = 0; r < 8; ++r) {
      const int mm = r + half * 8;              // C/D VGPR layout
      Y[(long long)(tile * 16 + mm) * NOUT + t * 16 + m] = acc[t][r];
    }
  }
}

// ----------------------------------------------------- edge gather/scatter
// lane-per-(edge,feature): 32 (or 16) consecutive lanes share one edge, so the
// gather of hin[src] and the atomic scatter to agg[dst] are both coalesced.
template<int F>
__global__ void k_edge_agg(const int* __restrict__ src, const int* __restrict__ dst,
                           const float* __restrict__ dinv,
                           const float* __restrict__ hin,
                           float* __restrict__ agg) {
  const unsigned idx = blockIdx.x * blockDim.x + threadIdx.x;
  const unsigned e = idx / F;
  const unsigned f = idx & (F - 1);
  if (e >= (unsigned)NE) return;
  const int s = src[e], d = dst[e];
  const float w = dinv[s] * dinv[d];
  atomicAdd(agg + (long long)d * F + f, hin[(long long)s * F + f] * w);
}

// self-loop term + bias (+ReLU) folded into one pass; writes agg in place
template<int F, bool RELU>
__global__ void k_finalize(float* __restrict__ agg, const float* __restrict__ hlin,
                           const float* __restrict__ dinv,
                           const float* __restrict__ bias) {
  const int i = blockIdx.x * blockDim.x + threadIdx.x;
  if (i >= NN * F) return;
  const int node = i / F;
  const int f    = i & (F - 1);
  const float di = dinv[node];
  float v = agg[i] + hlin[i] * di * di + bias[f];
  if (RELU) v = fmaxf(v, 0.0f);
  agg[i] = v;
}

// ---------------------------------------------------------------------------
extern "C" void kernel_launch(void* const* d_in, const int* in_sizes, int n_in,
                              void* d_out, int out_size, void* d_ws, size_t ws_size,
                              hipStream_t stream) {
  const int*   x    = (const int*)  d_in[0];            // [NN]
  const int*   ei   = (const int*)  d_in[1];            // [2, NE] flat
  const float* emb  = (const float*)d_in[2];            // [NN, DIN]
  const float* W1   = (const float*)d_in[3];            // [DIN, DHID]
  const float* b1   = (const float*)d_in[4];            // [DHID]
  const float* W2   = (const float*)d_in[5];            // [DHID, DOUT]
  const float* b2   = (const float*)d_in[6];            // [DOUT]
  float*       out  = (float*)d_out;                    // [NN, DOUT]

  const int* src = ei;
  const int* dst = ei + NE;

  // workspace layout (~33 MB): deg | dinv | h1lin | agg1(=h1 act) | h2lin
  float* ws    = (float*)d_ws;
  int*   deg   = (int*)ws;                 ws += NN;
  float* dinv  = ws;                       ws += NN;
  float* h1lin = ws;                       ws += (long long)NN * DHID;
  float* agg1  = ws;                       ws += (long long)NN * DHID;
  float* h2lin = ws;

  const int B = 256;
  auto blocks = [](long long n, int b) { return (int)((n + b - 1) / b); };

  // 0) zero-init accumulators (d_out/d_ws are poisoned; must be deterministic)
  k_zero_f32<<<blocks(NN, B), B, 0, stream>>>((float*)deg, NN);
  k_zero_f32<<<blocks((long long)NN * DHID, B), B, 0, stream>>>(agg1, NN * DHID);
  k_zero_f32<<<blocks((long long)NN * DOUT, B), B, 0, stream>>>(out, NN * DOUT);

  // 1) degree + symmetric normalization
  k_deg <<<blocks(NE, B), B, 0, stream>>>(dst, deg);
  k_dinv<<<blocks(NN, B), B, 0, stream>>>(deg, dinv);

  // 2) layer 1 dense: h1lin = emb[x] @ W1   (WMMA fp32)
  const int tiles = NN / 16;                       // 6250, exact
  const int wpb   = B / 32;                        // 8 waves per block
  k_linear_wmma<DIN, DHID, true><<<blocks(tiles, wpb), B, 0, stream>>>(emb, W1, x, h1lin);

  // 3) layer 1 edge aggregation (atomics into L2-resident agg1)
  k_edge_agg<DHID><<<blocks((long long)NE * DHID, B), B, 0, stream>>>(src, dst, dinv, h1lin, agg1);

  // 4) layer 1 finalize: self-loop + bias + ReLU (in place -> h1 activation)
  k_finalize<DHID, true><<<blocks((long long)NN * DHID, B), B, 0, stream>>>(agg1, h1lin, dinv, b1);

  // 5) layer 2 dense: h2lin = h1 @ W2   (WMMA fp32)
  k_linear_wmma<DHID, DOUT, false><<<blocks(tiles, wpb), B, 0, stream>>>(agg1, W2, nullptr, h2lin);

  // 6) layer 2 edge aggregation directly into d_out
  k_edge_agg<DOUT><<<blocks((long long)NE * DOUT, B), B, 0, stream>>>(src, dst, dinv, h2lin, out);

  // 7) layer 2 finalize: self-loop + bias (no ReLU)
  k_finalize<DOUT, false><<<blocks((long long)NN * DOUT, B), B, 0, stream>>>(out, h2lin, dinv, b2);
}